// PowerSpectrum_19593640804370
// MI455X (gfx1250) — compile-verified
//
#include <hip/hip_runtime.h>
#include <math.h>

typedef __attribute__((ext_vector_type(2))) float v2f;
typedef __attribute__((ext_vector_type(8))) float v8f;

#define NSAMPLES 2000
#define NGRAD    10000
#define NPROPS   32
#define NSOAP    (6 * 32 * 32)   // 6144

__device__ __forceinline__ v8f wmma4(v2f a, v2f b, v8f c) {
  // D = A(16x4 f32) * B(4x16 f32) + C(16x16 f32)
  return __builtin_amdgcn_wmma_f32_16x16x4_f32(false, a, false, b, (short)0, c,
                                               false, false);
}

// One wave computes the full 32x32 block  out = fact * v^T v  for one sample.
template <int L>
__global__ __launch_bounds__(256) void ps_values_kernel(
    const float* __restrict__ v, float* __restrict__ out, float fact) {
  constexpr int K   = 2 * L + 1;
  constexpr int NCH = (K + 3) / 4;

  const int wave  = blockIdx.x * (blockDim.x >> 5) + (threadIdx.x >> 5);
  const int lane  = threadIdx.x & 31;
  const int col   = lane & 15;
  const int hi    = lane >> 4;      // 0 or 1
  const int khalf = hi * 2;

  const float* __restrict__ vb = v + (size_t)wave * (K * NPROPS);

  v8f a00 = {}, a01 = {}, a10 = {}, a11 = {};
#pragma unroll
  for (int c = 0; c < NCH; ++c) {
    v2f f0, f1;
#pragma unroll
    for (int j = 0; j < 2; ++j) {
      const int  k  = c * 4 + j + khalf;
      const bool in = (k < K);
      f0[j] = in ? vb[k * NPROPS + col]      : 0.0f;
      f1[j] = in ? vb[k * NPROPS + 16 + col] : 0.0f;
    }
    a00 = wmma4(f0, f0, a00);
    a01 = wmma4(f0, f1, a01);
    a10 = wmma4(f1, f0, a10);
    a11 = wmma4(f1, f1, a11);
  }

  float* __restrict__ orow = out + (size_t)wave * NSOAP;
  const int mofs = hi * 8;
#pragma unroll
  for (int r = 0; r < 8; ++r) {
    const int m = r + mofs;
    orow[(m     ) * NPROPS + col     ] = a00[r] * fact;
    orow[(m     ) * NPROPS + col + 16] = a01[r] * fact;
    orow[(m + 16) * NPROPS + col     ] = a10[r] * fact;
    orow[(m + 16) * NPROPS + col + 16] = a11[r] * fact;
  }
}

// One wave per gradient sample ng; loops x = 0..2.
// out block = g^T (fact*v[idx]) + (fact*v[idx])^T g, two WMMA accumulations.
template <int L>
__global__ __launch_bounds__(256) void ps_grads_kernel(
    const float* __restrict__ v, const float* __restrict__ g,
    const int* __restrict__ idx, float* __restrict__ out, float fact) {
  constexpr int K   = 2 * L + 1;
  constexpr int NCH = (K + 3) / 4;

  const int ng    = blockIdx.x * (blockDim.x >> 5) + (threadIdx.x >> 5);
  const int lane  = threadIdx.x & 31;
  const int col   = lane & 15;
  const int hi    = lane >> 4;
  const int khalf = hi * 2;

  const int i = idx[ng];  // uniform per wave
  const float* __restrict__ vb = v + (size_t)i * (K * NPROPS);

  // Pre-load (and pre-scale) the gathered value fragments once; reuse for x=0..2.
  v2f vf0[NCH], vf1[NCH];
#pragma unroll
  for (int c = 0; c < NCH; ++c) {
#pragma unroll
    for (int j = 0; j < 2; ++j) {
      const int  k  = c * 4 + j + khalf;
      const bool in = (k < K);
      vf0[c][j] = in ? vb[k * NPROPS + col]      * fact : 0.0f;
      vf1[c][j] = in ? vb[k * NPROPS + 16 + col] * fact : 0.0f;
    }
  }

  for (int x = 0; x < 3; ++x) {
    const float* __restrict__ gb = g + ((size_t)ng * 3 + x) * (K * NPROPS);
    v8f a00 = {}, a01 = {}, a10 = {}, a11 = {};
#pragma unroll
    for (int c = 0; c < NCH; ++c) {
      v2f gf0, gf1;
#pragma unroll
      for (int j = 0; j < 2; ++j) {
        const int  k  = c * 4 + j + khalf;
        const bool in = (k < K);
        gf0[j] = in ? gb[k * NPROPS + col]      : 0.0f;
        gf1[j] = in ? gb[k * NPROPS + 16 + col] : 0.0f;
      }
      // term1: g^T vg   (A = g frag, B = vg frag)
      // term2: vg^T g   (A = vg frag, B = g frag)
      a00 = wmma4(gf0, vf0[c], a00);  a00 = wmma4(vf0[c], gf0, a00);
      a01 = wmma4(gf0, vf1[c], a01);  a01 = wmma4(vf0[c], gf1, a01);
      a10 = wmma4(gf1, vf0[c], a10);  a10 = wmma4(vf1[c], gf0, a10);
      a11 = wmma4(gf1, vf1[c], a11);  a11 = wmma4(vf1[c], gf1, a11);
    }

    float* __restrict__ orow = out + ((size_t)ng * 3 + x) * NSOAP;
    const int mofs = hi * 8;
#pragma unroll
    for (int r = 0; r < 8; ++r) {
      const int m = r + mofs;
      orow[(m     ) * NPROPS + col     ] = a00[r];
      orow[(m     ) * NPROPS + col + 16] = a01[r];
      orow[(m + 16) * NPROPS + col     ] = a10[r];
      orow[(m + 16) * NPROPS + col + 16] = a11[r];
    }
  }
}

extern "C" void kernel_launch(void* const* d_in, const int* in_sizes, int n_in,
                              void* d_out, int out_size, void* d_ws, size_t ws_size,
                              hipStream_t stream) {
  (void)in_sizes; (void)n_in; (void)out_size; (void)d_ws; (void)ws_size;
  // setup_inputs() dict order: values_0, grads_0, values_1, grads_1, ..., grad_sample_idx
  const float* values[6];
  const float* grads[6];
  for (int l = 0; l < 6; ++l) {
    values[l] = (const float*)d_in[2 * l];
    grads[l]  = (const float*)d_in[2 * l + 1];
  }
  const int* idx = (const int*)d_in[12];

  float* out  = (float*)d_out;
  float* outv = out;                                   // datafull [2000, 6144]
  float* outg = out + (size_t)NSAMPLES * NSOAP;        // datafull_grad [10000, 3, 6144]

  const dim3 blk(256);                                 // 8 waves per block
  const int vblocks = NSAMPLES / 8;                    // 250
  const int gblocks = NGRAD / 8;                       // 1250

#define PS_LAUNCH(L)                                                          \
  do {                                                                        \
    const float fact = 1.0f / sqrtf((float)(2 * (L) + 1));                    \
    ps_values_kernel<L><<<vblocks, blk, 0, stream>>>(values[L],               \
                                                     outv + (L) * 1024, fact);\
    ps_grads_kernel<L><<<gblocks, blk, 0, stream>>>(values[L], grads[L], idx, \
                                                    outg + (L) * 1024, fact); \
  } while (0)

  PS_LAUNCH(0);
  PS_LAUNCH(1);
  PS_LAUNCH(2);
  PS_LAUNCH(3);
  PS_LAUNCH(4);
  PS_LAUNCH(5);
#undef PS_LAUNCH
}